// TripletLoss_67439576482170
// MI455X (gfx1250) — compile-verified
//
#include <hip/hip_runtime.h>
#include <hip/hip_bf16.h>

typedef __attribute__((ext_vector_type(2))) float v2f;
typedef __attribute__((ext_vector_type(8))) float v8f;

#define TL_MARGIN 0.3f
#define TL_EPS 1e-6f

__global__ void tl_init(float* ws) {
    ws[0] = 0.0f;  // loss sum
    ws[1] = 0.0f;  // valid count
}

__global__ __launch_bounds__(128) void tl_main(
    const float* __restrict__ emb_t, const float* __restrict__ emb_t1,
    const int* __restrict__ ids_t, const int* __restrict__ ids_t1,
    float* __restrict__ ws, int M, int N, int totalWaves)
{
    constexpr int D  = 128;
    constexpr int KS = D / 4;        // 32 k-steps of K=4
    constexpr int KH = KS / 2;       // process B-frags in 2 chunks (VGPR pressure)

    const int lane = threadIdx.x & 31;
    const int wid  = blockIdx.x * (blockDim.x >> 5) + (threadIdx.x >> 5);
    if (wid >= totalWaves) return;   // uniform per wave -> EXEC all-ones inside

    const int mtiles = M >> 4;
    const int b  = wid / mtiles;
    const int m0 = (wid % mtiles) << 4;

    const int col = lane & 15;       // row-within-tile this lane loads / column of C
    const int hi  = lane >> 4;       // half-wave select (K pair / M half)

    // ---- preload A fragments: lane holds emb_t row (m0+col), floats {4s+2hi, 4s+2hi+1}
    const float* arow = emb_t + ((size_t)b * M + (m0 + col)) * D + 2 * hi;
    v2f aF[KS];
    float rN = 0.0f, rS = 0.0f;      // partial ||a||^2, sum(a) for row m0+col
#pragma unroll
    for (int s = 0; s < KS; ++s) {
        v2f a = *(const v2f*)(arow + 4 * s);
        aF[s] = a;
        rN += a.x * a.x + a.y * a.y;
        rS += a.x + a.y;
    }
    rN += __shfl_xor(rN, 16, 32);
    rS += __shfl_xor(rS, 16, 32);
    const int idOwn = ids_t[(size_t)b * M + m0 + col];

    // redistribute row stats to C-fragment layout: lane needs rows v + 8*hi
    float rowN[8], rowS[8];
    int   idm[8];
#pragma unroll
    for (int v = 0; v < 8; ++v) {
        const int src = v + 8 * hi;
        rowN[v] = __shfl(rN, src, 32);
        rowS[v] = __shfl(rS, src, 32);
        idm[v]  = __shfl(idOwn, src, 32);
    }

    const float BIGF = 1e30f;
    float runNeg[8], runPosD[8];
    int   runPosI[8];
#pragma unroll
    for (int v = 0; v < 8; ++v) { runNeg[v] = BIGF; runPosD[v] = 0.0f; runPosI[v] = 0x7fffffff; }

    const int ntiles = N >> 4;
    for (int nt = 0; nt < ntiles; ++nt) {
        const int n0 = nt << 4;
        const float* brow = emb_t1 + ((size_t)b * N + (n0 + col)) * D + 2 * hi;
        const int idn = ids_t1[(size_t)b * N + n0 + col];

        float cN = 0.0f, cS = 0.0f;  // partial ||b||^2, sum(b) for col n0+col
        v8f c = {0.f, 0.f, 0.f, 0.f, 0.f, 0.f, 0.f, 0.f};

        // two half-passes over K to cap live B-frag registers at 32 VGPRs
#pragma unroll
        for (int h = 0; h < 2; ++h) {
            v2f bF[KH];
#pragma unroll
            for (int s = 0; s < KH; ++s) {
                v2f bb = *(const v2f*)(brow + 4 * (h * KH + s));
                bF[s] = bb;
                cN += bb.x * bb.x + bb.y * bb.y;
                cS += bb.x + bb.y;
            }
#pragma unroll
            for (int s = 0; s < KH; ++s) {
                // D = A x B + C ; fp32 16x16x4 keeps full reference precision
                c = __builtin_amdgcn_wmma_f32_16x16x4_f32(
                        false, aF[h * KH + s], false, bF[s], (short)0, c, false, false);
            }
        }
        cN += __shfl_xor(cN, 16, 32);
        cS += __shfl_xor(cS, 16, 32);

        const int nCol = n0 + col;
#pragma unroll
        for (int v = 0; v < 8; ++v) {
            const float g  = c[v];
            float d2 = rowN[v] + cN - 2.0f * g
                     + 2.0f * TL_EPS * (rowS[v] - cS)
                     + (float)D * TL_EPS * TL_EPS;
            float d  = sqrtf(fmaxf(d2, 1e-12f));

            const bool same = (idn == idm[v]);
            int   pI = same ? nCol : 0x7fffffff;                 // first-positive candidate
            float pD = d;
            float nD = (!same && idn > 0) ? d : BIGF;            // hard-negative candidate

            // butterfly reduce across the 16 lanes of this half-wave
#pragma unroll
            for (int off = 1; off < 16; off <<= 1) {
                float oN = __shfl_xor(nD, off, 32);
                nD = fminf(nD, oN);
                int   oI = __shfl_xor(pI, off, 32);
                float oD = __shfl_xor(pD, off, 32);
                if (oI < pI) { pI = oI; pD = oD; }
            }
            runNeg[v] = fminf(runNeg[v], nD);
            if (pI < runPosI[v]) { runPosI[v] = pI; runPosD[v] = pD; }
        }
    }

    // lane 0 owns rows 0..7, lane 16 owns rows 8..15
    if (col == 0) {
        float lsum = 0.0f, lcnt = 0.0f;
#pragma unroll
        for (int v = 0; v < 8; ++v) {
            const bool valid = (idm[v] > 0) && (runPosI[v] != 0x7fffffff) && (runNeg[v] < BIGF);
            if (valid) {
                lsum += fmaxf(runPosD[v] - runNeg[v] + TL_MARGIN, 0.0f);
                lcnt += 1.0f;
            }
        }
        atomicAdd(&ws[0], lsum);
        atomicAdd(&ws[1], lcnt);
    }
}

__global__ void tl_final(const float* __restrict__ ws, float* __restrict__ out) {
    out[0] = ws[0] / fmaxf(ws[1], 1.0f);
}

extern "C" void kernel_launch(void* const* d_in, const int* in_sizes, int n_in,
                              void* d_out, int out_size, void* d_ws, size_t ws_size,
                              hipStream_t stream) {
    const float* emb_t  = (const float*)d_in[0];
    const float* emb_t1 = (const float*)d_in[1];
    const int*   ids_t  = (const int*)d_in[2];
    const int*   ids_t1 = (const int*)d_in[3];
    float* out = (float*)d_out;
    float* ws  = (float*)d_ws;

    const int B = 4;                       // fixed by reference setup
    const int M = in_sizes[2] / B;         // ids_t is [B, M]
    const int N = in_sizes[3] / B;         // ids_t1 is [B, N]

    const int totalWaves = B * (M >> 4);   // one wave per 16-anchor tile
    const int wavesPerBlk = 4;             // 128 threads = 4 wave32
    const int blocks = (totalWaves + wavesPerBlk - 1) / wavesPerBlk;

    tl_init<<<1, 1, 0, stream>>>(ws);
    tl_main<<<blocks, 32 * wavesPerBlk, 0, stream>>>(emb_t, emb_t1, ids_t, ids_t1,
                                                     ws, M, N, totalWaves);
    tl_final<<<1, 1, 0, stream>>>(ws, out);
}